// S4DKernel_41369124995193
// MI455X (gfx1250) — compile-verified
//
#include <hip/hip_runtime.h>

typedef __attribute__((ext_vector_type(2))) float v2f;
typedef __attribute__((ext_vector_type(8))) float v8f;

#define NH    32    // N/2 complex modes per head
#define HH    1024  // heads
#define LLEN  4096  // sequence length
#define LHALF 128   // a-range processed per half (a in [0,256))

struct cplx { float r, i; };
__device__ __forceinline__ cplx cmul(cplx a, cplx b) {
    return { a.r * b.r - a.i * b.i, a.r * b.i + a.i * b.r };
}

// One block = one head. 256 threads = 8 waves (wave32). Each wave owns one
// 16-row M-tile per half; 16 k-steps of V_WMMA_F32_16X16X4_F32 contract the
// 64-deep (real+imag x 32 modes) reduction. B-fragments (Q powers) are held
// in registers across both halves; Qi is stored pre-negated in LDS so the
// inner loop is pure {2x ds_load_2addr + v_wmma}.
__global__ __launch_bounds__(256) void s4d_wmma_kernel(
    const float* __restrict__ log_dt,
    const float* __restrict__ C,
    const float* __restrict__ log_A_real,
    const float* __restrict__ A_imag,
    float* __restrict__ out)
{
    __shared__ float sGr[NH][LHALF];   // Re(2*Ceff * w16^a)
    __shared__ float sGi[NH][LHALF];   // Im(...)
    __shared__ float sQr[NH][16];      // Re(w^b)
    __shared__ float sQmi[NH][16];     // -Im(w^b)   (pre-negated)
    __shared__ float swr[NH],    swi[NH];
    __shared__ float sw8r[NH],   sw8i[NH];
    __shared__ float sw16r[NH],  sw16i[NH];
    __shared__ float sw256r[NH], sw256i[NH];
    __shared__ float scer[NH],   scei[NH];

    const int h = blockIdx.x;
    const int t = threadIdx.x;

    // ---------------- Phase 0: per-mode parameters (threads 0..31) ----------
    if (t < NH) {
        const int n = t;
        const float dt = expf(log_dt[h]);
        const float Ar = -expf(log_A_real[h * NH + n]);
        const float Ai = A_imag[h * NH + n];
        const float xr = Ar * dt, xi = Ai * dt;
        const float ex = expf(xr);
        cplx w { ex * cosf(xi), ex * sinf(xi) };          // w = exp(dt*A)

        // Ceff2 = 2 * Cc * (w - 1) / A
        const float den = Ar * Ar + Ai * Ai;
        cplx invA { Ar / den, -Ai / den };
        cplx em1  { w.r - 1.0f, w.i };
        cplx tp   = cmul(em1, invA);
        cplx Cc   { C[h * 2 * NH + n], C[h * 2 * NH + NH + n] };
        cplx ce   = cmul(Cc, tp);
        ce.r *= 2.0f; ce.i *= 2.0f;

        cplx w2   = cmul(w, w);
        cplx w4   = cmul(w2, w2);
        cplx w8   = cmul(w4, w4);
        cplx w16  = cmul(w8, w8);
        cplx w32  = cmul(w16, w16);
        cplx w64  = cmul(w32, w32);
        cplx w128 = cmul(w64, w64);
        cplx w256 = cmul(w128, w128);

        swr[n] = w.r;       swi[n] = w.i;
        sw8r[n] = w8.r;     sw8i[n] = w8.i;
        sw16r[n] = w16.r;   sw16i[n] = w16.i;
        sw256r[n] = w256.r; sw256i[n] = w256.i;
        scer[n] = ce.r;     scei[n] = ce.i;
    }
    __syncthreads();

    // ---------------- Phase 1: Q table, Q[n][b] = w^b, b in [0,16) ----------
    {
        const int n = t & (NH - 1);
        const int b = t >> 5;                 // 0..7
        cplx w  { swr[n], swi[n] };
        cplx w2l = cmul(w, w);
        cplx w4l = cmul(w2l, w2l);
        cplx q { 1.0f, 0.0f };
        if (b & 1) q = cmul(q, w);
        if (b & 2) q = cmul(q, w2l);
        if (b & 4) q = cmul(q, w4l);
        sQr[n][b] = q.r;   sQmi[n][b] = -q.i;
        cplx w8 { sw8r[n], sw8i[n] };
        cplx q2 = cmul(q, w8);
        sQr[n][b + 8] = q2.r;  sQmi[n][b + 8] = -q2.i;
    }
    __syncthreads();

    const int lane = t & 31;
    const int wv   = t >> 5;          // wave id 0..7
    const int row  = lane & 15;       // A-matrix M index == B/D N index
    const int hiK  = (lane >= 16) ? 2 : 0;
    const int hiM  = (lane >= 16) ? 8 : 0;

    // ---------------- Hoist B fragments (invariant across halves) -----------
    v2f bq[16];
    #pragma unroll
    for (int s = 0; s < 16; ++s) {
        const int na = 4 * (s & 7) + hiK;
        const int nb = na + 1;
        if (s < 8) {
            bq[s].x = sQr[na][row];   bq[s].y = sQr[nb][row];
        } else {
            bq[s].x = sQmi[na][row];  bq[s].y = sQmi[nb][row];
        }
    }

    for (int half = 0; half < 2; ++half) {
        if (half) __syncthreads();    // prior half's WMMA reads complete

        // ------------ G build: G[n][a] = Ceff2[n] * w16[n]^a  ---------------
        {
            const int n = t >> 3;             // 0..31
            const int j = t & 7;              // 0..7 -> 16 consecutive a
            const int e = half * 8 + j;       // w256 exponent, 0..15
            cplx w256  { sw256r[n], sw256i[n] };
            cplx w512  = cmul(w256, w256);
            cplx w1024 = cmul(w512, w512);
            cplx w2048 = cmul(w1024, w1024);
            cplx p { 1.0f, 0.0f };
            if (e & 1) p = cmul(p, w256);
            if (e & 2) p = cmul(p, w512);
            if (e & 4) p = cmul(p, w1024);
            if (e & 8) p = cmul(p, w2048);
            cplx cur = cmul(cplx{ scer[n], scei[n] }, p);
            cplx w16 { sw16r[n], sw16i[n] };
            #pragma unroll
            for (int i2 = 0; i2 < 16; ++i2) {
                const int al = j * 16 + i2;
                sGr[n][al] = cur.r;
                sGi[n][al] = cur.i;
                cur = cmul(cur, w16);
            }
        }
        __syncthreads();

        // ------------ GEMM tile: D(16x16) += A(16x4) x B(4x16), 16 steps ----
        v8f acc = {};                          // C starts at zero
        const int aoff = wv * 16 + row;        // a index within this half
        #pragma unroll
        for (int s = 0; s < 16; ++s) {
            const int na = 4 * (s & 7) + hiK;  // mode index for VGPR0 slot
            const int nb = na + 1;             // mode index for VGPR1 slot
            v2f a;
            if (s < 8) {                       // real partial products
                a.x = sGr[na][aoff];  a.y = sGr[nb][aoff];
            } else {                           // imag partial products
                a.x = sGi[na][aoff];  a.y = sGi[nb][aoff];
            }
            acc = __builtin_amdgcn_wmma_f32_16x16x4_f32(
                /*neg_a=*/false, a, /*neg_b=*/false, bq[s],
                /*c_mod=*/(short)0, acc, /*reuse_a=*/false, /*reuse_b=*/false);
        }

        // ------------ Store D: l = 16*(half*128 + wv*16 + M) + N ------------
        const int abase = half * LHALF + wv * 16;
        #pragma unroll
        for (int r = 0; r < 8; ++r) {
            const int M = r + hiM;
            const int l = 16 * (abase + M) + row;
            out[h * LLEN + l] = acc[r];
        }
    }
}

extern "C" void kernel_launch(void* const* d_in, const int* in_sizes, int n_in,
                              void* d_out, int out_size, void* d_ws, size_t ws_size,
                              hipStream_t stream) {
    (void)in_sizes; (void)n_in; (void)out_size; (void)d_ws; (void)ws_size;
    const float* log_dt     = (const float*)d_in[0];
    const float* C          = (const float*)d_in[1];
    const float* log_A_real = (const float*)d_in[2];
    const float* A_imag     = (const float*)d_in[3];
    float* out = (float*)d_out;

    s4d_wmma_kernel<<<dim3(HH), dim3(256), 0, stream>>>(
        log_dt, C, log_A_real, A_imag, out);
}